// PointGather_78915729097542
// MI455X (gfx1250) — compile-verified
//
#include <hip/hip_runtime.h>
#include <hip/hip_bf16.h>

// PointGather for MI455X (gfx1250, wave32).
//
// Pure gather op, ~385 MB compulsory HBM traffic -> ~16.5 us at 23.3 TB/s.
// Roofline says memory-bound with zero matmul FLOPs, so no WMMA by design.
//
// MI455X-specific strategy (192 MB global L2):
//   * direct gather from range_features[B,C,H,W] (173.6 MB < 192 MB L2):
//     random per-point channel-column gathers become L2 hits after first
//     touch, so HBM sees only compulsory traffic — better than a transpose
//     pre-pass (+350 MB traffic).
//   * L2-residency protection: the write-once 190 MB output uses
//     non-temporal stores (TH_STORE_NT), and the read-once streams
//     points (13.6 MB) + ri_indices (5.4 MB) use non-temporal loads, so
//     neither rinses the gather table out of L2.
//
// Mapping: one wave (32 lanes) per point.
//   - wave-uniform loads: points[n*5] (batch idx), ri_indices[n] (b64), seg
//   - lane l gathers channels l and l+32 (stride H*W, saddr + lane voffset)
//   - fully coalesced stores of the 69-float output row + 1 float mask

#define PG_B   4
#define PG_H   64
#define PG_W   2650
#define PG_C   64
#define PG_HW  (PG_H * PG_W)       /* 169600 */
#define PG_THR 0.5f

__global__ __launch_bounds__(256)
void PointGather_78915729097542_kernel(const float* __restrict__ points,
                                       const int*   __restrict__ ri,
                                       const float* __restrict__ seg,
                                       const float* __restrict__ rf,
                                       float*       __restrict__ out,
                                       int N)
{
    const int wavesPerBlock = blockDim.x >> 5;
    const int n    = blockIdx.x * wavesPerBlock + (threadIdx.x >> 5); // point id
    const int lane = threadIdx.x & 31;
    if (n >= N) return;

    // ---- wave-uniform per-point scalars (broadcast loads, 1 request/wave).
    // points/ri are streamed exactly once -> NT loads to spare L2 capacity.
    const float p0 = __builtin_nontemporal_load(points + n * 5);  // batch idx
    const int   b  = (int)p0;
    const int   r  = __builtin_nontemporal_load(ri + n * 2 + 0);
    const int   c  = __builtin_nontemporal_load(ri + n * 2 + 1);

    const int pix = r * PG_W + c;            // 0 .. HW-1
    const float s = seg[b * PG_HW + pix];    // gathered seg score (L2 resident)
    const float m = (s >= PG_THR) ? 1.0f : 0.0f;

    // ---- per-lane channel gathers: ((b*C + ch)*HW + pix), ch = lane, lane+32.
    // Random across waves, lane*HW voffset within a wave; the 173.6 MB table
    // stays resident in the 192 MB L2.
    const float* col = rf + (b * PG_C) * PG_HW + pix;
    const float f0 = col[lane * PG_HW];
    const float f1 = col[(lane + 32) * PG_HW];

    // ---- write-once output: non-temporal so it doesn't rinse L2 ----
    float* orow = out + n * 69;
    __builtin_nontemporal_store(f0 * m, orow + 5 + lane);        // feats 0..31
    __builtin_nontemporal_store(f1 * m, orow + 37 + lane);       // feats 32..63
    if (lane > 0 && lane < 5) {
        const float pv = __builtin_nontemporal_load(points + n * 5 + lane);
        __builtin_nontemporal_store(pv * m, orow + lane);        // coords 1..4
    }
    if (lane == 0) {
        __builtin_nontemporal_store(p0 * m, orow);               // coord 0 (reuse p0)
        // tuple output #2: mask as float (1.0 / 0.0), flat after the N*69 block
        __builtin_nontemporal_store(m, out + N * 69 + n);
    }
}

extern "C" void kernel_launch(void* const* d_in, const int* in_sizes, int n_in,
                              void* d_out, int out_size, void* d_ws, size_t ws_size,
                              hipStream_t stream)
{
    const float* points = (const float*)d_in[0];   // [N,5]  float32
    const int*   ri     = (const int*)  d_in[1];   // [N,2]  int32
    const float* seg    = (const float*)d_in[2];   // [B,H,W] float32
    const float* rf     = (const float*)d_in[3];   // [B,C,H,W] float32
    float*       out    = (float*)d_out;           // [N,69] floats then [N] mask

    const int N = in_sizes[0] / 5;

    const int threads       = 256;                 // 8 wave32 per block
    const int wavesPerBlock = threads / 32;
    const int blocks        = (N + wavesPerBlock - 1) / wavesPerBlock;

    PointGather_78915729097542_kernel<<<blocks, threads, 0, stream>>>(
        points, ri, seg, rf, out, N);
}